// SelectiveSSM_90572270338477
// MI455X (gfx1250) — compile-verified
//
#include <hip/hip_runtime.h>
#include <hip/hip_bf16.h>

#define DMODEL   1024
#define DSTATE   16
#define DCONV    4
#define DINNER   2048
#define DTRANK   64
#define BSZ      2
#define SEQL     2048
#define NROWS    (BSZ * SEQL)                 // 4096
#define XPROJ_N  (DTRANK + 2 * DSTATE)        // 96

#define LDSPAD   20   // floats per LDS tile row: 16B-aligned (80B) & bank-conflict-free

typedef float v2f __attribute__((ext_vector_type(2)));
typedef float v8f __attribute__((ext_vector_type(8)));

// Async global->LDS copy of one 16-byte chunk per lane (CDNA5 TDM-lite path,
// tracked by ASYNCcnt; data never touches VGPRs).
__device__ __forceinline__ void async_b128_to_lds(const float* gbase, unsigned gByteOff,
                                                  void* ldsPtr)
{
    const unsigned ldsOff = (unsigned)(uintptr_t)ldsPtr;   // low 32 bits = LDS byte offset
    asm volatile("global_load_async_to_lds_b128 %0, %1, %2"
                 :: "v"(ldsOff), "v"(gByteOff),
                    "s"((unsigned long long)(uintptr_t)gbase)
                 : "memory");
}

// ---------------------------------------------------------------------------
// Tiled fp32 WMMA GEMM:  C[m,n] = sum_k A[m*lda + k] * W[n*ldw + k]
// 64x64 C tile / 128-thread block; 4 waves each own a 32x32 quadrant as
// 2x2 tiles of V_WMMA_F32_16X16X4_F32. K is streamed in 16-wide slabs via
// double-buffered LDS filled with GLOBAL_LOAD_ASYNC_TO_LDS_B128.
// ---------------------------------------------------------------------------
__global__ __launch_bounds__(128)
void gemm_f32_wmma(const float* __restrict__ A, const float* __restrict__ W,
                   float* __restrict__ C, int M, int N, int K,
                   int lda, int ldw, int ldc)
{
    __shared__ float As[2][64][LDSPAD];
    __shared__ float Ws[2][64][LDSPAD];

    const int tid  = threadIdx.x;
    const int lane = tid & 31;
    const int wave = tid >> 5;
    const int mBase = blockIdx.y * 64;
    const int nBase = blockIdx.x * 64;
    const int wm = (wave >> 1) * 32;   // wave row offset within 64x64 tile
    const int wn = (wave & 1) * 32;    // wave col offset

    v8f acc[2][2];
#pragma unroll
    for (int i = 0; i < 2; ++i)
#pragma unroll
        for (int j = 0; j < 2; ++j)
            acc[i][j] = (v8f){0.f, 0.f, 0.f, 0.f, 0.f, 0.f, 0.f, 0.f};

    const int lrow = tid >> 2;        // 0..31
    const int lcol = (tid & 3) * 4;   // 0,4,8,12
    const int lh   = lane & 15;
    const int ls   = lane >> 4;

    // Issue one K-slab fill (4 async instructions per thread). Out-of-range
    // rows are clamped to a valid address: their data only reaches C entries
    // that the store-time guard discards, so no predication is needed and
    // the fill runs with full EXEC.
    auto issue_stage = [&](int buf, int k0) {
#pragma unroll
        for (int rr = 0; rr < 2; ++rr) {
            const int row = lrow + rr * 32;
            int gm = mBase + row; if (gm >= M) gm = M - 1;
            async_b128_to_lds(A, (unsigned)(((size_t)gm * lda + k0 + lcol) * 4),
                              &As[buf][row][lcol]);
            int gn = nBase + row; if (gn >= N) gn = N - 1;
            async_b128_to_lds(W, (unsigned)(((size_t)gn * ldw + k0 + lcol) * 4),
                              &Ws[buf][row][lcol]);
        }
    };

    issue_stage(0, 0);

    for (int k0 = 0; k0 < K; k0 += 16) {
        const int buf = (k0 >> 4) & 1;
        if (k0 + 16 < K) {
            issue_stage(buf ^ 1, k0 + 16);
            // 4 next-slab asyncs outstanding allowed; in-order completion
            // means the current slab's 4 are drained.
            asm volatile("s_wait_asynccnt 0x4" ::: "memory");
        } else {
            asm volatile("s_wait_asynccnt 0x0" ::: "memory");
        }
        __syncthreads();   // all waves' fills for `buf` visible

#pragma unroll
        for (int kk = 0; kk < 16; kk += 4) {
            v2f af[2], bf[2];
#pragma unroll
            for (int i = 0; i < 2; ++i) {
                const float* ap = &As[buf][wm + i * 16 + lh][kk + 2 * ls];
                af[i] = (v2f){ap[0], ap[1]};
                const float* bp = &Ws[buf][wn + i * 16 + lh][kk + 2 * ls];
                bf[i] = (v2f){bp[0], bp[1]};
            }
#pragma unroll
            for (int i = 0; i < 2; ++i)
#pragma unroll
                for (int j = 0; j < 2; ++j)
                    acc[i][j] = __builtin_amdgcn_wmma_f32_16x16x4_f32(
                        false, af[i], false, bf[j],
                        (short)0, acc[i][j], false, false);
        }
        __syncthreads();   // nobody refills `buf` until every wave consumed it
    }

    // C/D layout: VGPR v -> m = v + 8*(lane>>4), n = lane&15
#pragma unroll
    for (int i = 0; i < 2; ++i)
#pragma unroll
        for (int j = 0; j < 2; ++j)
#pragma unroll
            for (int v = 0; v < 8; ++v) {
                const int gm = mBase + wm + i * 16 + v + 8 * ls;
                const int gn = nBase + wn + j * 16 + lh;
                if (gm < M && gn < N)
                    C[(size_t)gm * ldc + gn] = acc[i][j][v];
            }
}

// ---------------------------------------------------------------------------
// Causal depthwise conv (width 4) + bias + SiLU over the xc half of xz.
// ---------------------------------------------------------------------------
__global__ __launch_bounds__(256)
void conv_silu_kernel(const float* __restrict__ xz, const float* __restrict__ cw,
                      const float* __restrict__ cb, float* __restrict__ xc)
{
    const int id = blockIdx.x * blockDim.x + threadIdx.x;
    if (id >= NROWS * DINNER) return;
    const int d  = id & (DINNER - 1);
    const int bl = id >> 11;            // b*SEQL + l
    const int l  = bl & (SEQL - 1);
    const int b  = bl >> 11;

    float s = cb[d];
#pragma unroll
    for (int j = 0; j < DCONV; ++j) {
        const int ll = l - (DCONV - 1) + j;
        if (ll >= 0) {
            const float xv = xz[((size_t)(b * SEQL + ll)) * (2 * DINNER) + d];
            s += cw[d * DCONV + j] * xv;
        }
    }
    const float sig = 1.f / (1.f + __expf(-s));
    xc[(size_t)bl * DINNER + d] = s * sig;
}

// dt = softplus(dt_lin + bias[d]), in place
__global__ __launch_bounds__(256)
void bias_softplus_kernel(float* __restrict__ dt, const float* __restrict__ bias)
{
    const int id = blockIdx.x * blockDim.x + threadIdx.x;
    if (id >= NROWS * DINNER) return;
    const int d = id & (DINNER - 1);
    const float v = dt[id] + bias[d];
    dt[id] = (v > 20.f) ? v : log1pf(__expf(v));
}

// ---------------------------------------------------------------------------
// Selective scan: one lane per (b, d, n). 65536 lanes = 2048 wave32s.
// Serial chain per step: 1 exp + 2 fma + 4 shfl_xor (16-lane tree reduce).
// Working set < 192 MB -> per-step operands are L2 hits.
// ---------------------------------------------------------------------------
__global__ __launch_bounds__(256)
void selective_scan_kernel(const float* __restrict__ xc, const float* __restrict__ dt,
                           const float* __restrict__ x_dbl, const float* __restrict__ A_log,
                           float* __restrict__ y)
{
    const int id = blockIdx.x * blockDim.x + threadIdx.x;   // 0..65535
    const int n  = id & (DSTATE - 1);
    const int c  = id >> 4;                 // channel index 0..4095
    const int d  = c & (DINNER - 1);
    const int b  = c >> 11;

    const float A = -__expf(A_log[d * DSTATE + n]);
    float h = 0.f;
    const size_t rowBase = (size_t)b * SEQL;

    for (int l = 0; l < SEQL; ++l) {
        const size_t r   = rowBase + l;
        const float  dtv = dt[r * DINNER + d];
        const float  xv  = xc[r * DINNER + d];
        const float* xd  = x_dbl + r * XPROJ_N;
        const float  bv  = xd[DTRANK + n];
        const float  cv  = xd[DTRANK + DSTATE + n];

        const float dA = __expf(dtv * A);
        h = dA * h + dtv * bv * xv;

        float p = h * cv;
        p += __shfl_xor(p, 1);
        p += __shfl_xor(p, 2);
        p += __shfl_xor(p, 4);
        p += __shfl_xor(p, 8);
        if (n == 0) y[r * DINNER + d] = p;
    }
}

// y = (y + D[d]*xc) * silu(z), z = xz[..., DINNER + d]; in place into y
__global__ __launch_bounds__(256)
void gate_kernel(float* __restrict__ y, const float* __restrict__ xc,
                 const float* __restrict__ xz, const float* __restrict__ Dp)
{
    const int id = blockIdx.x * blockDim.x + threadIdx.x;
    if (id >= NROWS * DINNER) return;
    const int d  = id & (DINNER - 1);
    const size_t r = (size_t)(id >> 11);
    const float z = xz[r * (2 * DINNER) + DINNER + d];
    const float sig = 1.f / (1.f + __expf(-z));
    y[id] = (y[id] + Dp[d] * xc[id]) * (z * sig);
}

// ---------------------------------------------------------------------------
extern "C" void kernel_launch(void* const* d_in, const int* in_sizes, int n_in,
                              void* d_out, int out_size, void* d_ws, size_t ws_size,
                              hipStream_t stream)
{
    (void)in_sizes; (void)n_in; (void)out_size; (void)ws_size;

    const float* x          = (const float*)d_in[0];
    const float* in_proj_w  = (const float*)d_in[1];
    const float* conv_w     = (const float*)d_in[2];
    const float* conv_b     = (const float*)d_in[3];
    const float* A_log      = (const float*)d_in[4];
    const float* D_param    = (const float*)d_in[5];
    const float* x_proj_w   = (const float*)d_in[6];
    const float* dt_proj_w  = (const float*)d_in[7];
    const float* dt_proj_b  = (const float*)d_in[8];
    const float* out_proj_w = (const float*)d_in[9];
    float*       out        = (float*)d_out;

    // Workspace layout
    char* ws = (char*)d_ws;
    float* xz    = (float*)(ws);                                        // 64 MB
    float* xc    = (float*)(ws + (size_t)NROWS * 2 * DINNER * 4);       // 32 MB
    float* x_dbl = (float*)(ws + (size_t)NROWS * 3 * DINNER * 4);       // 1.5 MB
    float* dtf   = (float*)(ws + (size_t)NROWS * 3 * DINNER * 4
                               + (size_t)NROWS * XPROJ_N * 4);          // 32 MB
    float* y     = (float*)(ws + (size_t)NROWS * 4 * DINNER * 4
                               + (size_t)NROWS * XPROJ_N * 4);          // 32 MB

    const int nElem = NROWS * DINNER;                 // 8388608
    const int eBlocks = (nElem + 255) / 256;

    // 1. xz = x @ in_proj_w.T
    gemm_f32_wmma<<<dim3((2 * DINNER) / 64, NROWS / 64), 128, 0, stream>>>(
        x, in_proj_w, xz, NROWS, 2 * DINNER, DMODEL, DMODEL, DMODEL, 2 * DINNER);

    // 2. causal depthwise conv + SiLU -> xc
    conv_silu_kernel<<<eBlocks, 256, 0, stream>>>(xz, conv_w, conv_b, xc);

    // 3. x_dbl = xc @ x_proj_w.T
    gemm_f32_wmma<<<dim3((XPROJ_N + 63) / 64, NROWS / 64), 128, 0, stream>>>(
        xc, x_proj_w, x_dbl, NROWS, XPROJ_N, DINNER, DINNER, DINNER, XPROJ_N);

    // 4. dtf = x_dbl[:, :64] @ dt_proj_w.T   (K=64, lda=96)
    gemm_f32_wmma<<<dim3(DINNER / 64, NROWS / 64), 128, 0, stream>>>(
        x_dbl, dt_proj_w, dtf, NROWS, DINNER, DTRANK, XPROJ_N, DTRANK, DINNER);

    // 5. dtf = softplus(dtf + dt_proj_b)
    bias_softplus_kernel<<<eBlocks, 256, 0, stream>>>(dtf, dt_proj_b);

    // 6. selective scan -> y
    selective_scan_kernel<<<(NROWS * DSTATE) / 256, 256, 0, stream>>>(
        xc, dtf, x_dbl, A_log, y);

    // 7. y = (y + D*xc) * silu(z)
    gate_kernel<<<eBlocks, 256, 0, stream>>>(y, xc, xz, D_param);

    // 8. out = y @ out_proj_w.T
    gemm_f32_wmma<<<dim3(DMODEL / 64, NROWS / 64), 128, 0, stream>>>(
        y, out_proj_w, out, NROWS, DMODEL, DINNER, DINNER, DINNER, DMODEL);
}